// ChamferLoss_51986284151191
// MI455X (gfx1250) — compile-verified
//
#include <hip/hip_runtime.h>

// CDNA5 / gfx1250 Chamfer loss.
// dist(x,y) = |x|^2 + |y|^2 - 2 x.y ; cross term for a 16x16 tile pair is one
// V_WMMA_F32_16X16X4_F32 (K=4: 3 coords + zero pad) = 256 f32 dot products.
// Targets are staged through LDS in double-buffered 1024-point chunks so all
// 8 waves of the block share one global fetch (8x L2-traffic reduction);
// register-prefetch overlaps the next chunk's global loads with compute.

typedef float v2f __attribute__((ext_vector_type(2)));
typedef float v8f __attribute__((ext_vector_type(8)));

#define WAVES_PER_BLOCK 8
#define CHUNK_PTS 1024                        // target points per LDS chunk
#define F4_PER_CHUNK (CHUNK_PTS * 3 / 4)      // 768 float4 = 12 KB
#define F4_PER_THREAD (F4_PER_CHUNK / (WAVES_PER_BLOCK * 32)) // 3
#define TILES_PER_CHUNK (CHUNK_PTS / 16)      // 64

__global__ void __launch_bounds__(WAVES_PER_BLOCK * 32)
chamfer_dir_kernel(const float* __restrict__ src, const float* __restrict__ tgt,
                   int n_tgt, float* __restrict__ acc)
{
    __shared__ float4 lbuf[2][F4_PER_CHUNK];  // 24 KB of the WGP's 320 KB LDS

    const int tid  = threadIdx.x;
    const int lane = tid & 31;
    const int wave = tid >> 5;
    const int tile = blockIdx.x * WAVES_PER_BLOCK + wave; // source tile (16 pts)
    const int m    = lane & 15;
    const bool hi  = lane >= 16;

    // ---- A tile: 16 source points ----
    // A 16x4 f32 layout: lanes 0-15 -> row m, K=(0,1) in (a.x,a.y);
    //                    lanes 16-31 -> row m, K=(2,3) -> (z, 0).
    const float* sp = src + (size_t)(tile * 16 + m) * 3;
    const float sx = sp[0], sy = sp[1], sz = sp[2];
    v2f a;
    a.x = hi ? sz : sx;
    a.y = hi ? 0.0f : sy;
    const float pn = sx * sx + sy * sy + sz * sz; // ||src[m]||^2

    // Broadcast source norms to match C/D row layout: VGPR r holds M=r (low
    // half) or M=r+8 (high half).
    float prn[8];
#pragma unroll
    for (int r = 0; r < 8; ++r) {
        const float lo = __shfl(pn, r, 32);
        const float hv = __shfl(pn, r + 8, 32);
        prn[r] = hi ? hv : lo;
    }

    float minacc[8];
#pragma unroll
    for (int r = 0; r < 8; ++r) minacc[r] = 3.0e38f;

    const int nchunk = n_tgt / CHUNK_PTS;
    const float4* gt4 = (const float4*)tgt;   // chunk stride 12288 B: 16B aligned

    // Prefetch chunk 0 into registers.
    float4 pf[F4_PER_THREAD];
#pragma unroll
    for (int k = 0; k < F4_PER_THREAD; ++k)
        pf[k] = gt4[(size_t)k * (WAVES_PER_BLOCK * 32) + tid];

    int buf = 0;
    for (int c = 0; c < nchunk; ++c) {
        // Commit prefetched chunk c to LDS buffer `buf`.
#pragma unroll
        for (int k = 0; k < F4_PER_THREAD; ++k)
            lbuf[buf][k * (WAVES_PER_BLOCK * 32) + tid] = pf[k];

        // Start global loads for chunk c+1 (latency hidden by 64 tiles of math).
        if (c + 1 < nchunk) {
            const size_t base = (size_t)(c + 1) * F4_PER_CHUNK;
#pragma unroll
            for (int k = 0; k < F4_PER_THREAD; ++k)
                pf[k] = gt4[base + (size_t)k * (WAVES_PER_BLOCK * 32) + tid];
        }

        __syncthreads(); // stores to `buf` visible; also fences reuse (2-deep)

        const float* lf = (const float*)&lbuf[buf][0];
        for (int jj = 0; jj < TILES_PER_CHUNK; ++jj) {
            // ---- B tile from LDS: column n = lane&15; low half K=(0,1),
            // high half K=(2,3). Same-address reads across halves broadcast.
            const float* tp = lf + (size_t)(jj * 16 + m) * 3;
            const float tx = tp[0], ty = tp[1], tz = tp[2];
            v2f b;
            b.x = hi ? tz : tx;
            b.y = hi ? 0.0f : ty;
            const float tn = tx * tx + ty * ty + tz * tz; // ||tgt[n]||^2, lane-correct

            v8f cm = {};
            cm = __builtin_amdgcn_wmma_f32_16x16x4_f32(
                    /*neg_a=*/false, a, /*neg_b=*/false, b,
                    /*c_mod=*/(short)0, cm, /*reuse_a=*/false, /*reuse_b=*/false);

#pragma unroll
            for (int r = 0; r < 8; ++r) {
                const float d = prn[r] + tn - 2.0f * cm[r];
                minacc[r] = fminf(minacc[r], d);
            }
        }
        buf ^= 1;
    }

    // Min-reduce over N within each 16-lane half (masks 1,2,4,8 stay in-half).
#pragma unroll
    for (int r = 0; r < 8; ++r) {
        float v = minacc[r];
        v = fminf(v, __shfl_xor(v, 1, 32));
        v = fminf(v, __shfl_xor(v, 2, 32));
        v = fminf(v, __shfl_xor(v, 4, 32));
        v = fminf(v, __shfl_xor(v, 8, 32));
        minacc[r] = v;
    }
    float local = 0.0f;
#pragma unroll
    for (int r = 0; r < 8; ++r) local += minacc[r];
    // lane 0: sum of row-mins M=0..7 ; lane 16: M=8..15
    if (lane == 0 || lane == 16)
        atomicAdd(acc, local); // global_atomic_add_f32
}

__global__ void chamfer_init_kernel(float* acc)
{
    acc[0] = 0.0f;
    acc[1] = 0.0f;
}

__global__ void chamfer_finalize_kernel(const float* __restrict__ acc,
                                        float* __restrict__ out,
                                        float inv_np, float inv_nt)
{
    out[0] = acc[0] * inv_np + acc[1] * inv_nt;
}

extern "C" void kernel_launch(void* const* d_in, const int* in_sizes, int n_in,
                              void* d_out, int out_size, void* d_ws, size_t ws_size,
                              hipStream_t stream)
{
    const float* pred   = (const float*)d_in[0];
    const float* target = (const float*)d_in[1];
    const int n_pred = in_sizes[0] / 3; // 32768
    const int n_tgt  = in_sizes[1] / 3; // 32768
    float* acc = (float*)d_ws;
    float* out = (float*)d_out;

    chamfer_init_kernel<<<1, 1, 0, stream>>>(acc);

    const dim3 block(WAVES_PER_BLOCK * 32);
    const int grid_p = (n_pred / 16) / WAVES_PER_BLOCK; // 256 blocks
    const int grid_t = (n_tgt / 16) / WAVES_PER_BLOCK;  // 256 blocks

    // pred -> nearest target
    chamfer_dir_kernel<<<grid_p, block, 0, stream>>>(pred, target, n_tgt, acc + 0);
    // target -> nearest pred
    chamfer_dir_kernel<<<grid_t, block, 0, stream>>>(target, pred, n_pred, acc + 1);

    chamfer_finalize_kernel<<<1, 1, 0, stream>>>(acc, out,
                                                 1.0f / (float)n_pred,
                                                 1.0f / (float)n_tgt);
}